// ClassificationHead_18846316495193
// MI455X (gfx1250) — compile-verified
//
#include <hip/hip_runtime.h>
#include <hip/hip_bf16.h>

typedef __attribute__((ext_vector_type(2))) float v2f;
typedef __attribute__((ext_vector_type(8))) float v8f;

#define LN_EPS 1e-5f

// x: (256, 257, 768) f32. Effective rows: r in [0,65536), maps to
// source row (b*257 + n + 1) with b=r>>8, n=r&255.
// One wave (32 lanes) handles 16 consecutive output rows.
// WMMA f32 16x16x4: A = 16x4 tile of x (lane L holds rows L&15,
// K-pair 2*(L>>4)+{0,1} -> contiguous b64 load). B = g-chunk replicated
// across all 16 columns (same contiguous-pair lane layout) so every
// column of D accumulates Sg[m] = sum_k x[m,k]*g[k].
__global__ __launch_bounds__(256) void head_ln_sigmoid_kernel(
    const float* __restrict__ x,
    const float* __restrict__ ln_gamma,
    const float* __restrict__ ln_beta,
    const float* __restrict__ W,
    const float* __restrict__ bias,
    float* __restrict__ out)
{
    __shared__ __align__(16) float g_lds[768];
    __shared__ float redG[256];
    __shared__ float redB[256];
    __shared__ float sG, sBc;

    const int tid = threadIdx.x;

    // ---- per-block prep: g = gamma*W into LDS; G = sum g; Bc = sum beta*W + bias
    float pG = 0.0f, pB = 0.0f;
    #pragma unroll
    for (int i = 0; i < 3; ++i) {
        const int e = tid + i * 256;
        const float gv = ln_gamma[e] * W[e];
        g_lds[e] = gv;
        pG += gv;
        pB += ln_beta[e] * W[e];
    }
    redG[tid] = pG;
    redB[tid] = pB;
    __syncthreads();
    #pragma unroll
    for (int s = 128; s > 0; s >>= 1) {
        if (tid < s) { redG[tid] += redG[tid + s]; redB[tid] += redB[tid + s]; }
        __syncthreads();
    }
    if (tid == 0) { sG = redG[0]; sBc = redB[0] + bias[0]; }
    __syncthreads();
    const float G  = sG;
    const float Bc = sBc;

    // ---- main streaming loop: 16 rows per wave, K = 768 in 192 chunks of 4
    const int lane = tid & 31;
    const int wave = tid >> 5;
    const int tile = blockIdx.x * 8 + wave;      // 512 blocks * 8 waves = 4096 tiles
    const int m    = lane & 15;
    const int h    = lane >> 4;                  // K-half within chunk
    const int r    = tile * 16 + m;              // output row 0..65535
    const int bIdx = r >> 8;
    const int nIdx = r & 255;
    const float* __restrict__ xrow = x + (size_t)(bIdx * 257 + nIdx + 1) * 768 + 2 * h;

    v8f acc = {};                                // Sg accumulator (16x16 D tile)
    float s1a = 0.0f, s1b = 0.0f;                // per-lane S1 partials
    float s2a = 0.0f, s2b = 0.0f;                // per-lane S2 partials

    #pragma unroll 8
    for (int t = 0; t < 192; ++t) {
        const v2f xv = *(const v2f*)(xrow + 4 * t);              // global_load_b64
        const v2f gv = *(const v2f*)&g_lds[4 * t + 2 * h];       // ds_load_b64
        acc = __builtin_amdgcn_wmma_f32_16x16x4_f32(
            /*neg_a=*/false, xv, /*neg_b=*/false, gv,
            /*c_mod=*/(short)0, acc, /*reuse_a=*/false, /*reuse_b=*/false);
        s1a += xv.x;
        s1b += xv.y;
        s2a = fmaf(xv.x, xv.x, s2a);
        s2b = fmaf(xv.y, xv.y, s2b);
    }

    // combine the two K-halves (lane and lane+16) for S1/S2
    float s1 = s1a + s1b;
    float s2 = s2a + s2b;
    s1 += __shfl_xor(s1, 16, 32);
    s2 += __shfl_xor(s2, 16, 32);

    // Sg extraction: columns of D are replicated, so low lanes hold rows 0..7
    // in acc[0..7] and high lanes hold rows 8..15. Select acc[lane&7], then
    // pull from the correct half.
    const int k = lane & 7;
    float pick = acc[0];
    if (k == 1) pick = acc[1];
    if (k == 2) pick = acc[2];
    if (k == 3) pick = acc[3];
    if (k == 4) pick = acc[4];
    if (k == 5) pick = acc[5];
    if (k == 6) pick = acc[6];
    if (k == 7) pick = acc[7];
    const int src = (lane & 8) ? (lane + 16) : lane;
    const float sg = __shfl(pick, src, 32);

    if (lane < 16) {
        const float invE = 1.0f / 768.0f;
        const float mu    = s1 * invE;
        const float var   = fmaf(-mu, mu, s2 * invE);
        const float rinv  = rsqrtf(var + LN_EPS);
        const float logit = fmaf(rinv, sg - mu * G, Bc);
        out[r] = 1.0f / (1.0f + __expf(-logit));
    }
}

extern "C" void kernel_launch(void* const* d_in, const int* in_sizes, int n_in,
                              void* d_out, int out_size, void* d_ws, size_t ws_size,
                              hipStream_t stream) {
    const float* x        = (const float*)d_in[0];
    const float* ln_gamma = (const float*)d_in[1];
    const float* ln_beta  = (const float*)d_in[2];
    const float* W        = (const float*)d_in[3];
    const float* bias     = (const float*)d_in[4];
    float* out = (float*)d_out;

    // 65536 rows / 16 rows-per-wave = 4096 waves = 512 blocks of 8 waves
    head_ln_sigmoid_kernel<<<512, 256, 0, stream>>>(x, ln_gamma, ln_beta, W, bias, out);

    (void)in_sizes; (void)n_in; (void)out_size; (void)d_ws; (void)ws_size;
}